// SerialExpert_23648089932473
// MI455X (gfx1250) — compile-verified
//
#include <hip/hip_runtime.h>

// Problem constants (from reference): A[M,N] x W[E,N,K] -> out[M,E,K], fp32.
#define M_DIM 8192
#define N_DIM 1024
#define K_DIM 32
#define E_DIM 64
#define NCHUNK 64
#define NITER (N_DIM / NCHUNK)   // 16

typedef __attribute__((ext_vector_type(2))) float v2f;
typedef __attribute__((ext_vector_type(8))) float v8f;

// Block: 1024 threads = 32 waves (wave32).
// Each block covers 4 M-tiles x 4 experts x 2 k-tiles = 32 16x16 output tiles,
// one per wave. Software pipeline:
//   - A chunk (64x64 f32): async DMA to LDS (global_load_async_to_lds_b128),
//     double-buffered, chunk c+1 in flight during compute of chunk c.
//   - W chunk (4e x 64n x 32k): global float4 prefetch into registers one
//     chunk ahead; transposed ds-stores into single-buffered LDS.
// LDS pitch 68 floats -> per-step WMMA fragment reads are conflict-free
// ds_load_b64 across all 64 banks.
__global__ __launch_bounds__(1024) void moe_gemm_f32_wmma(
    const float* __restrict__ A,
    const float* __restrict__ W,
    float* __restrict__ out) {
  __shared__ float As[2][4][16][68];  // [buf][m_tile][row][n]   (2 x 17408 B)
  __shared__ float Bs[4][32][68];     // [expert][k_col][n] transposed (34816 B)

  const int t    = threadIdx.x;        // 0..1023
  const int lane = t & 31;
  const int wid  = t >> 5;             // 0..31
  const int kt   = wid & 1;            // k-tile -> output cols kt*16..+15
  const int el   = (wid >> 1) & 3;     // expert within block
  const int mt   = (wid >> 3) & 3;     // m-tile within block
  const int half = lane >> 4;          // 0: K={0,1}, 1: K={2,3} (ISA layout)
  const int lr   = lane & 15;

  const int m0 = blockIdx.x * 64;      // 128 blocks in x
  const int e0 = blockIdx.y * 4;       // 16 blocks in y

  // Staging indices.
  const int arow = t >> 4;             // 0..63: row within 64-row A chunk
  const int acol = (t & 15) << 2;      // 0..60: col within 64-col A chunk

  // Per-thread W-prefetch mapping: two float4 = (expert, n, k4).
  const int f0  = t * 2;
  const int bk4 = f0 & 7;              // float4 index along K (0..7)
  const int bn  = (f0 >> 3) & 63;      // n within chunk
  const int be  = f0 >> 9;             // expert 0..3
  const float* wsrc = &W[(size_t)(e0 + be) * (N_DIM * K_DIM)
                         + (size_t)bn * K_DIM + (bk4 << 2)];

  const float* asrc = A + (size_t)(m0 + arow) * N_DIM + acol;

  v8f acc = {0.f, 0.f, 0.f, 0.f, 0.f, 0.f, 0.f, 0.f};

  // ---- Prologue: async-DMA A chunk 0 into As[0]; prefetch W chunk 0 ----
  {
    unsigned lds = (unsigned)(uintptr_t)&As[0][arow >> 4][arow & 15][acol];
    unsigned long long ga = (unsigned long long)(uintptr_t)asrc;
    asm volatile("global_load_async_to_lds_b128 %0, %1, off"
                 :: "v"(lds), "v"(ga) : "memory");
  }
  float4 breg0 = *(const float4*)(wsrc);
  float4 breg1 = *(const float4*)(wsrc + K_DIM);  // second float4: n -> n+1? no: f+1 -> k4+1 unless wrap

  // NOTE: f0 is even and bk4 = f0&7 in 0..6 steps of 2? f0=2t -> bk4 even.
  // f0+1 has k4 = bk4+1, same (e,n): contiguous +4 floats.
  breg1 = *(const float4*)(wsrc + 4);

  for (int c = 0; c < NITER; ++c) {
    const int cur = c & 1;
    const int n1  = (c + 1) * NCHUNK;

    // Issue next A chunk DMA into the other buffer (overlaps with compute).
    if (c + 1 < NITER) {
      unsigned lds = (unsigned)(uintptr_t)&As[cur ^ 1][arow >> 4][arow & 15][acol];
      unsigned long long ga = (unsigned long long)(uintptr_t)(asrc + n1);
      asm volatile("global_load_async_to_lds_b128 %0, %1, off"
                   :: "v"(lds), "v"(ga) : "memory");
    }

    // Store prefetched W chunk c into LDS (transposed), then prefetch c+1.
    {
      Bs[be][bk4 * 4 + 0][bn] = breg0.x;
      Bs[be][bk4 * 4 + 1][bn] = breg0.y;
      Bs[be][bk4 * 4 + 2][bn] = breg0.z;
      Bs[be][bk4 * 4 + 3][bn] = breg0.w;
      Bs[be][bk4 * 4 + 4][bn] = breg1.x;
      Bs[be][bk4 * 4 + 5][bn] = breg1.y;
      Bs[be][bk4 * 4 + 6][bn] = breg1.z;
      Bs[be][bk4 * 4 + 7][bn] = breg1.w;
    }
    if (c + 1 < NITER) {
      const float* ws = wsrc + (size_t)n1 * K_DIM;
      breg0 = *(const float4*)(ws);
      breg1 = *(const float4*)(ws + 4);
      // Current A chunk's DMA is the older of 2 in flight (per-wave in-order).
      asm volatile("s_wait_asynccnt 0x1" ::: "memory");
    } else {
      asm volatile("s_wait_asynccnt 0x0" ::: "memory");
    }

    __syncthreads();  // As[cur] DMA-complete everywhere; Bs stores visible

    // ---- 16 WMMA steps of K=4 over this 64-wide chunk ----
    #pragma unroll
    for (int k = 0; k < 16; ++k) {
      const int nb = 4 * k + 2 * half;
      v2f a = *(const v2f*)&As[cur][mt][lr][nb];            // A frag 16x4
      v2f b = *(const v2f*)&Bs[el][kt * 16 + lr][nb];       // B frag 4x16
      acc = __builtin_amdgcn_wmma_f32_16x16x4_f32(
          /*neg_a=*/false, a, /*neg_b=*/false, b,
          /*c_mod=*/(short)0, acc, /*reuse_a=*/false, /*reuse_b=*/false);
    }

    __syncthreads();  // all waves done reading before buffers are rewritten
  }

  // ---- Store 16x16 f32 tile: VGPR j -> row m0+mt*16+j+8*half ----
  const int eg = e0 + el;
  float* op = out + (size_t)(m0 + mt * 16 + 8 * half) * (E_DIM * K_DIM)
                  + eg * K_DIM + kt * 16 + lr;
  #pragma unroll
  for (int j = 0; j < 8; ++j) {
    op[(size_t)j * (E_DIM * K_DIM)] = acc[j];
  }
}

extern "C" void kernel_launch(void* const* d_in, const int* in_sizes, int n_in,
                              void* d_out, int out_size, void* d_ws, size_t ws_size,
                              hipStream_t stream) {
  (void)in_sizes; (void)n_in; (void)out_size; (void)d_ws; (void)ws_size;
  const float* A = (const float*)d_in[0];   // [M, N]
  const float* W = (const float*)d_in[1];   // [E, N, K]
  float* out = (float*)d_out;               // [M, E, K]

  dim3 grid(M_DIM / 64, E_DIM / 4);         // 128 x 16 blocks
  dim3 block(1024);
  moe_gemm_f32_wmma<<<grid, block, 0, stream>>>(A, W, out);
}